// DoublePendulum_29807073034425
// MI455X (gfx1250) — compile-verified
//
#include <hip/hip_runtime.h>
#include <hip/hip_bf16.h>

// Double pendulum batch ODE integration (Tsit5, 4 fixed substeps per saveat).
// One thread per pendulum; state in registers; ts grid staged in LDS;
// output streamed with 128-bit non-temporal stores (256 MB write-once > 192 MB L2).
// Latency-bound (1024 waves, serial RK chain) -> hardware v_rcp_f32 instead of
// full IEEE div expansion shortens every stage's dependent chain.

typedef float v4f __attribute__((ext_vector_type(4)));

struct S4 { float a, b, c, d; };

#define FIELDS(X) X(a) X(b) X(c) X(d)

__device__ __forceinline__ S4 vf(const S4 y, const float gL) {
    // y = (th1, th2, w1, w2)
    const float dth = y.a - y.b;
    const float sd  = __sinf(dth);
    const float cd  = __cosf(dth);
    const float a1  = 0.5f * cd;            // a2 = 2*a1 = cd
    const float f1  = fmaf(-0.5f * y.d * y.d, sd, -gL * __sinf(y.a));
    const float f2  = fmaf( y.c * y.c,        sd, -gL * __sinf(y.b));
    // den = 1 - 0.5*cd^2 in [0.5, 1]: safe for 1-ulp hardware reciprocal.
    const float inv = __builtin_amdgcn_rcpf(fmaf(-a1, cd, 1.0f));
    S4 k;
    k.a = y.c;
    k.b = y.d;
    k.c = (f1 - a1 * f2) * inv;
    k.d = (f2 - cd * f1) * inv;
    return k;
}

__device__ __forceinline__ S4 tsit5_step(S4 y, const float h, const float gL) {
    constexpr float C21 = 0.161f;
    constexpr float C31 = -0.008480655492356989f, C32 = 0.335480655492357f;
    constexpr float C41 = 2.8971530571054935f,  C42 = -6.359448489975075f,
                    C43 = 4.3622954328695815f;
    constexpr float C51 = 5.325864828439257f,   C52 = -11.748883564062828f,
                    C53 = 7.4955393428898365f,  C54 = -0.09249506636175525f;
    constexpr float C61 = 5.86145544294642f,    C62 = -12.92096931784711f,
                    C63 = 8.159367898576159f,   C64 = -0.071584973281401f,
                    C65 = -0.028269050394068383f;
    constexpr float B1 = 0.09646076681806523f,  B2 = 0.01f,
                    B3 = 0.4798896504144996f,   B4 = 1.379008574103742f,
                    B5 = -3.290069515436081f,   B6 = 2.324710524099774f;

    const S4 k1 = vf(y, gL);

    S4 t2;
    const float hC21 = h * C21;
#define ST2(f) t2.f = fmaf(hC21, k1.f, y.f);
    FIELDS(ST2)
#undef ST2
    const S4 k2 = vf(t2, gL);

    S4 t3;
#define ST3(f) t3.f = fmaf(h, fmaf(C32, k2.f, C31 * k1.f), y.f);
    FIELDS(ST3)
#undef ST3
    const S4 k3 = vf(t3, gL);

    S4 t4;
#define ST4(f) t4.f = fmaf(h, fmaf(C43, k3.f, fmaf(C42, k2.f, C41 * k1.f)), y.f);
    FIELDS(ST4)
#undef ST4
    const S4 k4 = vf(t4, gL);

    S4 t5;
#define ST5(f) t5.f = fmaf(h, fmaf(C54, k4.f, fmaf(C53, k3.f, \
                      fmaf(C52, k2.f, C51 * k1.f))), y.f);
    FIELDS(ST5)
#undef ST5
    const S4 k5 = vf(t5, gL);

    S4 t6;
#define ST6(f) t6.f = fmaf(h, fmaf(C65, k5.f, fmaf(C64, k4.f, \
                      fmaf(C63, k3.f, fmaf(C62, k2.f, C61 * k1.f)))), y.f);
    FIELDS(ST6)
#undef ST6
    const S4 k6 = vf(t6, gL);

#define FIN(f) y.f = fmaf(h, fmaf(B6, k6.f, fmaf(B5, k5.f, fmaf(B4, k4.f, \
                      fmaf(B3, k3.f, fmaf(B2, k2.f, B1 * k1.f))))), y.f);
    FIELDS(FIN)
#undef FIN
    return y;
}

__global__ void __launch_bounds__(256)
DoublePendulum_29807073034425_kernel(const float* __restrict__ th1,
                                     const float* __restrict__ th2,
                                     const float* __restrict__ w1,
                                     const float* __restrict__ w2,
                                     const float* __restrict__ Lv,
                                     const float* __restrict__ ts,
                                     float* __restrict__ out,
                                     int B, int T) {
    extern __shared__ float s_ts[];
    // Stage the time grid (2 KB) into LDS once per block.
    for (int i = threadIdx.x; i < T; i += blockDim.x) s_ts[i] = ts[i];
    __syncthreads();

    const int b = blockIdx.x * blockDim.x + (int)threadIdx.x;
    if (b >= B) return;

    // gfx1250 prefetch of the per-pendulum state cachelines.
    __builtin_prefetch(th1 + b, 0, 3);
    __builtin_prefetch(th2 + b, 0, 3);
    __builtin_prefetch(w1  + b, 0, 3);
    __builtin_prefetch(w2  + b, 0, 3);
    __builtin_prefetch(Lv  + b, 0, 3);

    S4 y;
    y.a = th1[b];
    y.b = th2[b];
    y.c = w1[b];
    y.d = w2[b];
    const float gL = 9.8f / Lv[b];

    v4f* __restrict__ o = (v4f*)out;  // [T, B] of float4; lane-contiguous b128 stores

    // saveat includes t0
    {
        v4f v0 = {y.a, y.b, y.c, y.d};
        __builtin_nontemporal_store(v0, o + (size_t)b);
    }

    for (int t = 0; t < T - 1; ++t) {
        const float h = 0.25f * (s_ts[t + 1] - s_ts[t]);  // dt / SUBSTEPS
#pragma unroll
        for (int s = 0; s < 4; ++s) {
            y = tsit5_step(y, h, gL);
        }
        v4f v = {y.a, y.b, y.c, y.d};
        __builtin_nontemporal_store(v, o + (size_t)(t + 1) * (size_t)B + (size_t)b);
    }
}

extern "C" void kernel_launch(void* const* d_in, const int* in_sizes, int n_in,
                              void* d_out, int out_size, void* d_ws, size_t ws_size,
                              hipStream_t stream) {
    const float* th1 = (const float*)d_in[0];
    const float* th2 = (const float*)d_in[1];
    const float* w1  = (const float*)d_in[2];
    const float* w2  = (const float*)d_in[3];
    const float* Lv  = (const float*)d_in[4];
    const float* ts  = (const float*)d_in[5];
    float* out = (float*)d_out;

    const int B = in_sizes[0];
    const int T = in_sizes[5];

    const int block = 256;
    const int grid = (B + block - 1) / block;
    const size_t lds_bytes = (size_t)T * sizeof(float);

    DoublePendulum_29807073034425_kernel<<<grid, block, lds_bytes, stream>>>(
        th1, th2, w1, w2, Lv, ts, out, B, T);
}